// QuantumConvolutionLayer_69853348102899
// MI455X (gfx1250) — compile-verified
//
#include <hip/hip_runtime.h>
#include <stdint.h>

// ---------------------------------------------------------------------------
// Quanvolution layer for MI455X (gfx1250).
//
// Bandwidth-bound: ~67 MB total traffic -> ~2.9 us at 23.3 TB/s. Per patch
// (j2,k2) the two needed input rows are one contiguous 2 KB chunk, staged
// into LDS with the Tensor Data Mover (tensor_load_to_lds), double-buffered
// per wave and synchronized with s_wait_tensorcnt. Compute per element is a
// double-angle-reduced bilinear form: 4 transcendentals + 18 FMAs.
// ---------------------------------------------------------------------------

#define NPATCH 16384   // 128 x 128 patches
#define PPW    4       // patches per wave (double-buffered pipeline)

typedef unsigned int u32x4 __attribute__((ext_vector_type(4)));
typedef int          i32x8 __attribute__((ext_vector_type(8)));
typedef int          i32x4 __attribute__((ext_vector_type(4)));

// ------------------------- setup: 18 coefficients --------------------------
struct C2 { float re, im; };
__device__ static inline C2 cmul(C2 a, C2 b) {
  return C2{a.re * b.re - a.im * b.im, a.re * b.im + a.im * b.re};
}

__global__ void quanv_build_coeffs(const float* __restrict__ wts,
                                   float* __restrict__ Cout)
{
  if (threadIdx.x != 0 || blockIdx.x != 0) return;

  C2 U[4][4];
  for (int i = 0; i < 4; ++i)
    for (int j = 0; j < 4; ++j)
      U[i][j] = C2{(i == j) ? 1.0f : 0.0f, 0.0f};

  // Fixed-structure RandomLayers(seed=42) gate list (data table, editable).
  // type: 0=RX 1=RY 2=RZ 3=CNOT (w0=ctrl, w1=tgt). Params indexed (layer, i).
  const int NOPS     = 5;
  const int ty[5]    = {1, 2, 3, 1, 0};
  const int pl[5]    = {0, 0, 0, 1, 1};
  const int pidx[5]  = {0, 1, 0, 0, 1};
  const int w0a[5]   = {0, 1, 0, 1, 0};
  const int w1a[5]   = {0, 0, 1, 0, 0};

  for (int o = 0; o < NOPS; ++o) {
    C2 G[4][4];
    for (int i = 0; i < 4; ++i)
      for (int j = 0; j < 4; ++j) G[i][j] = C2{0.f, 0.f};

    if (ty[o] == 3) {                       // CNOT permutation
      for (int s = 0; s < 4; ++s) {
        int b0 = (s >> 1) & 1, b1 = s & 1;  // qubit0 = high bit
        int ctrl = (w0a[o] == 0) ? b0 : b1;
        int nb0 = b0, nb1 = b1;
        if (ctrl) { if (w1a[o] == 0) nb0 ^= 1; else nb1 ^= 1; }
        G[(nb0 << 1) | nb1][s] = C2{1.f, 0.f};
      }
    } else {                                // single-qubit rotation
      float th = wts[pl[o] * 2 + pidx[o]];
      float c = cosf(0.5f * th), s = sinf(0.5f * th);
      C2 g[2][2];
      if (ty[o] == 0) {                     // RX
        g[0][0] = C2{c, 0.f};  g[0][1] = C2{0.f, -s};
        g[1][0] = C2{0.f, -s}; g[1][1] = C2{c, 0.f};
      } else if (ty[o] == 1) {              // RY
        g[0][0] = C2{c, 0.f};  g[0][1] = C2{-s, 0.f};
        g[1][0] = C2{s, 0.f};  g[1][1] = C2{c, 0.f};
      } else {                              // RZ
        g[0][0] = C2{c, -s};   g[0][1] = C2{0.f, 0.f};
        g[1][0] = C2{0.f, 0.f}; g[1][1] = C2{c, s};
      }
      const int wire = w0a[o];
      for (int p = 0; p < 2; ++p)
        for (int q = 0; q < 2; ++q)
          for (int r = 0; r < 2; ++r)
            for (int s2 = 0; s2 < 2; ++s2) {
              C2 v = C2{0.f, 0.f};
              if (wire == 0) { if (q == s2) v = g[p][r]; }
              else           { if (p == r)  v = g[q][s2]; }
              G[2 * p + q][2 * r + s2] = v;
            }
    }
    // U = G * U
    C2 NU[4][4];
    for (int i = 0; i < 4; ++i)
      for (int j = 0; j < 4; ++j) {
        C2 acc{0.f, 0.f};
        for (int k = 0; k < 4; ++k) {
          C2 t = cmul(G[i][k], U[k][j]);
          acc.re += t.re; acc.im += t.im;
        }
        NU[i][j] = acc;
      }
    for (int i = 0; i < 4; ++i)
      for (int j = 0; j < 4; ++j) U[i][j] = NU[i][j];
  }

  // M_q = sum_u z_q[u] (ReU_u ReU_u^T + ImU_u ImU_u^T)
  const float z[2][4] = {{1.f, 1.f, -1.f, -1.f}, {1.f, -1.f, 1.f, -1.f}};
  float M[2][4][4];
  for (int q = 0; q < 2; ++q)
    for (int v = 0; v < 4; ++v)
      for (int w2 = 0; w2 < 4; ++w2) {
        float acc = 0.f;
        for (int u = 0; u < 4; ++u)
          acc += z[q][u] * (U[u][v].re * U[u][w2].re + U[u][v].im * U[u][w2].im);
        M[q][v][w2] = acc;
      }

  // Quadratic form -> trig basis: out_q = sum_{m,n} C_q[m][n] f_m(pi*phi0) f_n(pi*phi1)
  // with f = (1, cos, sin).  A_p A_r = a0 + a1*cos(2a) + a2*sin(2a):
  auto alf = [](int m, int p, int r) -> float {
    if (m == 0) return (p == r) ? 0.5f : 0.0f;
    if (m == 1) return (p == r) ? ((p == 0) ? 0.5f : -0.5f) : 0.0f;
    return (p != r) ? 0.5f : 0.0f;
  };
  for (int q = 0; q < 2; ++q)
    for (int m = 0; m < 3; ++m)
      for (int n = 0; n < 3; ++n) {
        float acc = 0.f;
        for (int p = 0; p < 2; ++p)
          for (int qq = 0; qq < 2; ++qq)
            for (int r = 0; r < 2; ++r)
              for (int ss = 0; ss < 2; ++ss)
                acc += M[q][2 * p + qq][2 * r + ss] * alf(m, p, r) * alf(n, qq, ss);
        Cout[q * 9 + m * 3 + n] = acc;
      }
}

// ---------------------- TDM 1-D tile load (2 KB/patch) ----------------------
__device__ static inline void tdm_issue_load(const float* x, uint32_t ldsOff, int patch)
{
  const int j2 = patch >> 7, k2 = patch & 127;
  int js = 2 * j2 + 1; if (js > 254) js = 254;
  int ks = 2 * k2 + 1; if (ks > 254) ks = 254;
  // phi0 row and phi1 row are contiguous: 512 floats starting here.
  const uint64_t ga = (uint64_t)(uintptr_t)x +
                      ((uint64_t)js * 65536u + (uint64_t)ks * 256u) * 4u;

  // D# group 0: count=1 | lds_addr | global_addr[56:0] | type=2
  u32x4 g0 = { 1u,
               ldsOff,
               (uint32_t)ga,
               (uint32_t)(ga >> 32) | (2u << 30) };
  // D# group 1: data_size=4B, tensor_dim0=512, tensor_dim1=1,
  //             tile_dim0=512, tile_dim1=1, tensor_dim0_stride=512
  i32x8 g1 = { (int)(2u << 16),
               (int)(512u << 16),
               (int)(1u << 16),
               (int)(512u << 16),
               1,
               512,
               0, 0 };
  // D# groups 2/3: higher dims = 1 (benign), tile dims unused
  i32x4 g2 = { 1, 1, 0, 0 };
  i32x4 g3 = { 0, (int)(1u << 16), 0, 0 };
  // 6-arg toolchain form: extra int32x8 group (unused for <=2D tile) + cpol
  i32x8 g4 = { 0, 0, 0, 0, 0, 0, 0, 0 };
  __builtin_amdgcn_tensor_load_to_lds(g0, g1, g2, g3, g4, 0);
}

// ------------------------------ main kernel --------------------------------
__global__ __launch_bounds__(32) void quanv_main(const float* __restrict__ x,
                                                 const float* __restrict__ C,
                                                 float* __restrict__ out)
{
  __shared__ float smem[2][512];       // double buffer: 2 x 2 KB
  const int lane = threadIdx.x;
  const int p0   = blockIdx.x * PPW;

  float c0m[9], c1m[9];
#pragma unroll
  for (int i = 0; i < 9; ++i) { c0m[i] = C[i]; c1m[i] = C[9 + i]; }

  const uint32_t ldsBase = (uint32_t)(uintptr_t)&smem[0][0];

  tdm_issue_load(x, ldsBase, p0);                       // prologue
  for (int t = 0; t < PPW; ++t) {
    const int p = p0 + t;
    const int slot = t & 1;
    if (t + 1 < PPW) {
      tdm_issue_load(x, ldsBase + (uint32_t)(((t + 1) & 1) * 2048), p + 1);
      __builtin_amdgcn_s_wait_tensorcnt(1);             // current tile landed
    } else {
      __builtin_amdgcn_s_wait_tensorcnt(0);
    }
    asm volatile("" ::: "memory");

    const int j2 = p >> 7, k2 = p & 127;
    const int obase = j2 * 65536 + k2 * 512;
    const float* ph = &smem[slot][0];
#pragma unroll
    for (int tt = 0; tt < 8; ++tt) {
      const int b = lane + 32 * tt;
      const float a0 = 3.14159265358979f * ph[b];        // 2*halfangle
      const float a1 = 3.14159265358979f * ph[256 + b];
      const float ca = __cosf(a0), sa = __sinf(a0);
      const float cb = __cosf(a1), sb = __sinf(a1);
      const float r0 = (c0m[0] + c0m[1] * cb + c0m[2] * sb)
                     + (c0m[3] + c0m[4] * cb + c0m[5] * sb) * ca
                     + (c0m[6] + c0m[7] * cb + c0m[8] * sb) * sa;
      const float r1 = (c1m[0] + c1m[1] * cb + c1m[2] * sb)
                     + (c1m[3] + c1m[4] * cb + c1m[5] * sb) * ca
                     + (c1m[6] + c1m[7] * cb + c1m[8] * sb) * sa;
      __builtin_nontemporal_store(r0, &out[obase + b]);        // q = 0
      __builtin_nontemporal_store(r1, &out[obase + 256 + b]);  // q = 1
    }
    // drain our LDS reads before the next TDM overwrites this slot
    asm volatile("s_wait_dscnt 0" ::: "memory");
  }
}

// ------------------------------ entry point --------------------------------
extern "C" void kernel_launch(void* const* d_in, const int* in_sizes, int n_in,
                              void* d_out, int out_size, void* d_ws, size_t ws_size,
                              hipStream_t stream)
{
  (void)in_sizes; (void)n_in; (void)out_size; (void)ws_size;
  const float* x   = (const float*)d_in[0];   // (256,256,256) f32
  const float* wts = (const float*)d_in[1];   // (2,2) f32
  float* out = (float*)d_out;                 // (128,128,2,256) f32
  float* C   = (float*)d_ws;                  // 18 coefficient floats

  quanv_build_coeffs<<<1, 1, 0, stream>>>(wts, C);
  quanv_main<<<NPATCH / PPW, 32, 0, stream>>>(x, C, out);
}